// CARCell_28441273434126
// MI455X (gfx1250) — compile-verified
//
#include <hip/hip_runtime.h>

// CARFAC CAR cell forward: B=32 batch, T=8192 steps, C=84 cascade channels.
// One wave32 per batch element; 3 channels per lane (28 active lanes).
// The channel cascade x <- g_c*(x+py_c) is an affine scan; g_c is constant in
// time, so all multiplicative (A) prefix products are hoisted out of the time
// loop and the per-step scan is B-only via DPP row-shifts (pure VALU latency).

#define NP    84
#define BATCH 32
#define TLEN  8192
#define CPL   3          // channels per lane
#define PI_F  3.14159265358979323846f

// DPP row_shr:N within 16-lane rows; invalid lanes receive `oldv` (identity).
#define DPP_SHR_F(oldv, v, ctrl)                                            \
    __int_as_float(__builtin_amdgcn_update_dpp(                             \
        __float_as_int(oldv), __float_as_int(v), (ctrl), 0xF, 0xF, false))

__device__ __forceinline__ float readlane_f(float v, int l) {
    return __int_as_float(__builtin_amdgcn_readlane(__float_as_int(v), l));
}

__global__ __launch_bounds__(32, 1)
void car_forward_kernel(const float* __restrict__ audio,   // [B,T]
                        const float* __restrict__ u0,      // [B,C]
                        const float* __restrict__ v0,      // [B,C]
                        const float* __restrict__ pv0,     // [B,C]
                        const float* __restrict__ p_hfdc,
                        const float* __restrict__ p_zero_ratio,
                        const float* __restrict__ p_min_zeta,
                        const float* __restrict__ p_max_zeta,
                        const float* __restrict__ p_erb_break,
                        const float* __restrict__ p_erb_q,
                        const float* __restrict__ p_v_offset,
                        const float* __restrict__ p_vel_scale,
                        float* __restrict__ out)           // [B,T,C]
{
    const int b    = blockIdx.x;
    const int lane = threadIdx.x;

    const float hfdc       = p_hfdc[0];
    const float zero_ratio = p_zero_ratio[0];
    const float min_zeta   = p_min_zeta[0];
    const float max_zeta   = p_max_zeta[0];
    const float erb_break  = p_erb_break[0];
    const float erb_q      = p_erb_q[0];
    const float v_offset   = p_v_offset[0];
    const float vel_scale  = p_vel_scale[0];

    const float f_ratio   = zero_ratio * zero_ratio - 1.0f;
    const float inv_erb_q = 1.0f / erb_q;

    // --- pole frequencies: exact sequential ERB descent (matches reference) ---
    float pf[CPL];
    {
        float f = 20400.0f;
        for (int k = 0; k < NP; ++k) {
            int rel = k - lane * CPL;
            if (rel >= 0 && rel < CPL) pf[rel] = f;
            f = f - 0.5f * (erb_break + f) * inv_erb_q;
        }
    }

    // --- per-channel constants ---
    float A0[CPL], C0[CPL], R1[CPL], ZR[CPL], G[CPL], GH[CPL];
    #pragma unroll
    for (int j = 0; j < CPL; ++j) {
        int c = lane * CPL + j;
        if (c < NP) {
            float f     = pf[j];
            float x     = f * (2.0f / 48000.0f);
            float theta = x * PI_F;
            float a0    = cosf(theta);
            float c0    = sinf(theta);
            float zr1   = PI_F * (x - hfdc * x * x * x);
            float r1    = 1.0f - zr1 * max_zeta;
            float erbpf = (erb_break + f) * inv_erb_q;
            float minz  = min_zeta + 0.25f * (erbpf / f - min_zeta);
            float zr    = zr1 * (max_zeta - minz);
            float rpz   = r1 + zr;
            float h     = c0 * f_ratio;
            float num   = 1.0f - 2.0f * rpz * a0 + rpz * rpz;
            float den   = num + h * rpz * c0;
            float g     = num / den;
            A0[j] = a0; C0[j] = c0; R1[j] = r1; ZR[j] = zr;
            G[j] = g; GH[j] = g * h;
        } else {
            // padding channel: exact identity in the cascade, zero state math
            A0[j] = 0.0f; C0[j] = 0.0f; R1[j] = 0.0f; ZR[j] = 0.0f;
            G[j]  = 1.0f; GH[j] = 0.0f;
        }
    }

    const bool upper = (lane >= 16);

    // --- time-invariant A-scan prefix products (hoisted out of time loop) ---
    // As_k = lane's accumulated A entering Kogge-Stone step k (B-combine coef)
    float As0 = G[2] * (G[1] * G[0]);
    float Ap;
    Ap = DPP_SHR_F(1.0f, As0, 0x111); float As1  = As0 * Ap;
    Ap = DPP_SHR_F(1.0f, As1, 0x112); float As2  = As1 * Ap;
    Ap = DPP_SHR_F(1.0f, As2, 0x114); float As3  = As2 * Ap;
    Ap = DPP_SHR_F(1.0f, As3, 0x118); float Arow = As3 * Ap;  // row inclusive
    float A15r  = readlane_f(Arow, 15);                       // row-0 total
    float AeRow = DPP_SHR_F(1.0f, Arow, 0x111);               // row exclusive
    float AeFull = upper ? AeRow * A15r : AeRow;              // wave exclusive

    // --- per-lane states ---
    float U[CPL], V[CPL], PV[CPL];
    #pragma unroll
    for (int j = 0; j < CPL; ++j) {
        int c = lane * CPL + j;
        if (c < NP) {
            U[j]  = u0 [b * NP + c];
            V[j]  = v0 [b * NP + c];
            PV[j] = pv0[b * NP + c];
        } else { U[j] = 0.0f; V[j] = 0.0f; PV[j] = 0.0f; }
    }

    const float* aud    = audio + (size_t)b * TLEN;
    float*       outb   = out   + (size_t)b * TLEN * NP;
    const bool   active = (lane * CPL < NP);   // lanes 0..27 write output

    float nxt = aud[0];

    for (int t = 0; t < TLEN; ++t) {
        float smp = nxt;
        int tn = (t + 1 < TLEN) ? (t + 1) : (TLEN - 1);
        nxt = aud[tn];                       // prefetch next sample early

        // --- per-channel resonator update (fully parallel, 3-way ILP) ---
        float PU[CPL], Bc[CPL];              // Bc = g*py (affine offset)
        #pragma unroll
        for (int j = 0; j < CPL; ++j) {
            float vel = V[j] - PV[j];
            float q   = fmaf(vel, vel_scale, v_offset);
            float nlf = __builtin_amdgcn_rcpf(fmaf(q, q, 1.0f));
            float r   = fmaf(ZR[j], nlf, R1[j]);
            float ru  = r * U[j];
            float rv  = r * V[j];
            PU[j]     = fmaf(A0[j], ru, -(C0[j] * rv));
            float vn  = fmaf(C0[j], ru, A0[j] * rv);
            Bc[j]     = GH[j] * vn;          // b_j = (g_j*h_j) * v_new
            PV[j] = V[j]; V[j] = vn;
        }

        // --- lane-local composed affine offset ---
        float Bi = fmaf(G[2], fmaf(G[1], Bc[0], Bc[1]), Bc[2]);

        // --- B-only Kogge-Stone scan (A coefs precomputed), DPP row_shr ---
        float Bp;
        Bp = DPP_SHR_F(0.0f, Bi, 0x111); Bi = fmaf(As0, Bp, Bi);
        Bp = DPP_SHR_F(0.0f, Bi, 0x112); Bi = fmaf(As1, Bp, Bi);
        Bp = DPP_SHR_F(0.0f, Bi, 0x114); Bi = fmaf(As2, Bp, Bi);
        Bp = DPP_SHR_F(0.0f, Bi, 0x118); Bi = fmaf(As3, Bp, Bi);
        float B15 = readlane_f(Bi, 15);                 // row-0 total offset
        float Be  = DPP_SHR_F(0.0f, Bi, 0x111);         // row exclusive
        float Bup = fmaf(AeRow, B15, Be);               // fold row-0 total
        Be = upper ? Bup : Be;

        // --- apply: x entering this lane's first channel ---
        float x = fmaf(AeFull, smp, Be);
        float y0, y1, y2;
        U[0] = PU[0] + x;  x = fmaf(G[0], x, Bc[0]);  y0 = x;
        U[1] = PU[1] + x;  x = fmaf(G[1], x, Bc[1]);  y1 = x;
        U[2] = PU[2] + x;  x = fmaf(G[2], x, Bc[2]);  y2 = x;

        if (active) {
            float* o = outb + (size_t)t * NP + lane * CPL;
            o[0] = y0; o[1] = y1; o[2] = y2;   // contiguous 12B/lane -> b96
        }
    }
}

extern "C" void kernel_launch(void* const* d_in, const int* in_sizes, int n_in,
                              void* d_out, int out_size, void* d_ws, size_t ws_size,
                              hipStream_t stream) {
    (void)in_sizes; (void)n_in; (void)d_ws; (void)ws_size; (void)out_size;
    const float* audio      = (const float*)d_in[0];
    const float* u0         = (const float*)d_in[1];
    const float* v0         = (const float*)d_in[2];
    const float* pv0        = (const float*)d_in[3];
    const float* hfdc       = (const float*)d_in[4];
    const float* zero_ratio = (const float*)d_in[5];
    const float* min_zeta   = (const float*)d_in[6];
    const float* max_zeta   = (const float*)d_in[7];
    const float* erb_break  = (const float*)d_in[8];
    const float* erb_q      = (const float*)d_in[9];
    const float* v_offset   = (const float*)d_in[10];
    const float* vel_scale  = (const float*)d_in[11];
    float* out = (float*)d_out;

    car_forward_kernel<<<dim3(BATCH), dim3(32), 0, stream>>>(
        audio, u0, v0, pv0, hfdc, zero_ratio, min_zeta, max_zeta,
        erb_break, erb_q, v_offset, vel_scale, out);
}